// GCN_26422638805210
// MI455X (gfx1250) — compile-verified
//
#include <hip/hip_runtime.h>
#include <cstdint>
#include <cstddef>

typedef __attribute__((ext_vector_type(2))) float v2f;
typedef __attribute__((ext_vector_type(8))) float v8f;

#define BN_EPS 1e-5f

// ---------------------------------------------------------------- utilities

__global__ void zero_f32(float* __restrict__ p, size_t n) {
  size_t i = (size_t)blockIdx.x * blockDim.x + threadIdx.x;
  size_t stride = (size_t)gridDim.x * blockDim.x;
  for (; i < n; i += stride) p[i] = 0.0f;
}

// ------------------------------------------------------- degree + scatter

__global__ void degree_kernel(const long long* __restrict__ dst,
                              float* __restrict__ deg, int E) {
  int i = blockIdx.x * blockDim.x + threadIdx.x;
  int stride = gridDim.x * blockDim.x;
  for (; i < E; i += stride) atomicAdd(&deg[(int)dst[i]], 1.0f);
}

__global__ void invdeg_kernel(float* __restrict__ deg, int n) {
  int i = blockIdx.x * blockDim.x + threadIdx.x;
  if (i < n) deg[i] = 1.0f / fmaxf(deg[i], 1.0f);
}

// agg[dst, f] += x[src, f]   (one thread per (edge, feature); F = 1<<fshift)
__global__ void scatter_feats(const float* __restrict__ x,
                              const long long* __restrict__ src,
                              const long long* __restrict__ dst,
                              float* __restrict__ agg,
                              size_t total, int fshift) {
  const int fmask = (1 << fshift) - 1;
  size_t i = (size_t)blockIdx.x * blockDim.x + threadIdx.x;
  size_t stride = (size_t)gridDim.x * blockDim.x;
  for (; i < total; i += stride) {
    size_t e = i >> fshift;
    int f = (int)(i & fmask);
    int s = (int)src[e];
    int d = (int)dst[e];
    atomicAdd(&agg[((size_t)d << fshift) + f], x[((size_t)s << fshift) + f]);
  }
}

// ------------------------------------------------------------- WMMA GEMM
// out[N, Fout] = (agg * invdeg[row]) @ Wl^T + xin @ Wr^T + bias
// Wl, Wr are [Fout, Fin] row-major (PyTorch linear weight layout).
// One 16x16 output tile per wave32; K stepped by 4 via V_WMMA_F32_16X16X4_F32.
__global__ void sage_gemm_wmma(const float* __restrict__ agg,
                               const float* __restrict__ invdeg,
                               const float* __restrict__ xin,
                               const float* __restrict__ Wl,
                               const float* __restrict__ Wr,
                               const float* __restrict__ bias,
                               float* __restrict__ out,
                               int Nn, int Fin, int Fout) {
  const int lane = threadIdx.x & 31;
  const int wave = threadIdx.x >> 5;
  const int wavesPerBlock = blockDim.x >> 5;
  const int tile = blockIdx.x * wavesPerBlock + wave;
  const int tiles_n = Fout >> 4;
  const int tm = tile / tiles_n;
  const int tn = tile - tm * tiles_n;
  if (tm * 16 >= Nn) return;            // wave-uniform: EXEC stays all-ones

  const int m0 = tm << 4;
  const int n0 = tn << 4;
  const int arow = m0 + (lane & 15);    // A rows striped over lanes 0-15 / 16-31
  const int bcol = n0 + (lane & 15);    // B cols likewise
  const int ksel = (lane >> 4) << 1;    // lanes 0-15 -> K+{0,1}; 16-31 -> K+{2,3}

  const float* __restrict__ aRow = agg + (size_t)arow * Fin;
  const float* __restrict__ xRow = xin + (size_t)arow * Fin;
  const float* __restrict__ wlRow = Wl + (size_t)bcol * Fin;
  const float* __restrict__ wrRow = Wr + (size_t)bcol * Fin;
  const float dscale = invdeg[arow];

  v8f c = {0.f, 0.f, 0.f, 0.f, 0.f, 0.f, 0.f, 0.f};

  // mean @ Wl^T
  for (int k0 = 0; k0 < Fin; k0 += 4) {
    v2f a, b;
    a.x = aRow[k0 + ksel + 0] * dscale;
    a.y = aRow[k0 + ksel + 1] * dscale;
    b.x = wlRow[k0 + ksel + 0];
    b.y = wlRow[k0 + ksel + 1];
    c = __builtin_amdgcn_wmma_f32_16x16x4_f32(false, a, false, b,
                                              (short)0, c, false, false);
  }
  // x @ Wr^T (accumulate into same C)
  for (int k0 = 0; k0 < Fin; k0 += 4) {
    v2f a, b;
    a.x = xRow[k0 + ksel + 0];
    a.y = xRow[k0 + ksel + 1];
    b.x = wrRow[k0 + ksel + 0];
    b.y = wrRow[k0 + ksel + 1];
    c = __builtin_amdgcn_wmma_f32_16x16x4_f32(false, a, false, b,
                                              (short)0, c, false, false);
  }

  // C/D layout: VGPR r -> row m0 + r (+8 for upper lane half), col n0 + (lane&15)
  const int rbase = m0 + ((lane >> 4) << 3);
  const int col = n0 + (lane & 15);
  const float bv = bias[col];
#pragma unroll
  for (int r = 0; r < 8; ++r) {
    out[(size_t)(rbase + r) * Fout + col] = c[r] + bv;
  }
}

// --------------------------------------------------- batch-norm statistics
// blockDim.x == F; thread j accumulates column j over a row-stripe, coalesced.
__global__ void col_stats(const float* __restrict__ x,
                          float* __restrict__ sum, float* __restrict__ sumsq,
                          int Nn, int F) {
  const int j = threadIdx.x;
  float s = 0.f, s2 = 0.f;
  for (int n = blockIdx.x; n < Nn; n += gridDim.x) {
    float v = x[(size_t)n * F + j];
    s += v;
    s2 += v * v;
  }
  atomicAdd(&sum[j], s);
  atomicAdd(&sumsq[j], s2);
}

__global__ void bn_relu(const float* __restrict__ xin, float* __restrict__ xout,
                        const float* __restrict__ sum, const float* __restrict__ sumsq,
                        const float* __restrict__ gamma, const float* __restrict__ beta,
                        int Nn, int F) {
  const float invN = 1.0f / (float)Nn;
  size_t total = (size_t)Nn * F;
  size_t i = (size_t)blockIdx.x * blockDim.x + threadIdx.x;
  size_t stride = (size_t)gridDim.x * blockDim.x;
  for (; i < total; i += stride) {
    int j = (int)(i % F);
    float mu = sum[j] * invN;
    float var = sumsq[j] * invN - mu * mu;
    float y = gamma[j] * (xin[i] - mu) * rsqrtf(var + BN_EPS) + beta[j];
    xout[i] = fmaxf(y, 0.0f);
  }
}

// --------------------------------------------------------- FC (64 -> 2) + softmax
__global__ void fc_softmax(const float* __restrict__ x0,
                           const float* __restrict__ Wfc,
                           const float* __restrict__ bfc,
                           float* __restrict__ x1, int Nn) {
  int n = blockIdx.x * blockDim.x + threadIdx.x;
  if (n >= Nn) return;
  const float* __restrict__ xr = x0 + (size_t)n * 64;
  float a0 = bfc[0], a1 = bfc[1];
#pragma unroll 8
  for (int k = 0; k < 64; ++k) {
    float v = xr[k];
    a0 += v * Wfc[k];
    a1 += v * Wfc[64 + k];
  }
  float m = fmaxf(a0, a1);
  float e0 = __expf(a0 - m), e1 = __expf(a1 - m);
  float inv = 1.0f / (e0 + e1);
  x1[(size_t)n * 2 + 0] = e0 * inv;
  x1[(size_t)n * 2 + 1] = e1 * inv;
}

// ------------------------------------------------------------------ driver

extern "C" void kernel_launch(void* const* d_in, const int* in_sizes, int n_in,
                              void* d_out, int out_size, void* d_ws, size_t ws_size,
                              hipStream_t stream) {
  const float*     x    = (const float*)d_in[0];
  const long long* ei   = (const long long*)d_in[1];   // int64 edge_index [2, E]
  const float*     Wl1  = (const float*)d_in[2];
  const float*     bl1  = (const float*)d_in[3];
  const float*     Wr1  = (const float*)d_in[4];
  const float*     g1   = (const float*)d_in[5];
  const float*     b1   = (const float*)d_in[6];
  const float*     Wl2  = (const float*)d_in[7];
  const float*     bl2  = (const float*)d_in[8];
  const float*     Wr2  = (const float*)d_in[9];
  const float*     g2   = (const float*)d_in[10];
  const float*     b2   = (const float*)d_in[11];
  const float*     Wfc  = (const float*)d_in[12];
  const float*     bfc  = (const float*)d_in[13];

  const int N = in_sizes[0] / 64;
  const int E = in_sizes[1] / 2;
  const long long* srcI = ei;
  const long long* dstI = ei + E;

  float* out_x0 = (float*)d_out;                       // [N, 64]
  float* out_x1 = (float*)d_out + (size_t)N * 64;      // [N, 2]

  // workspace layout (floats)
  float* ws    = (float*)d_ws;
  float* deg   = ws;                                   // N        (becomes invdeg)
  float* agg1  = deg  + N;                             // N*64
  float* h     = agg1 + (size_t)N * 64;                // N*128    (raw, then BN'd in place)
  float* agg2  = h    + (size_t)N * 128;               // N*128
  float* x0raw = agg2 + (size_t)N * 128;               // N*64
  float* stat  = x0raw + (size_t)N * 64;               // 256 (sum | sumsq)

  const int T = 256;
  auto blocks = [](size_t work, int t, size_t cap) {
    size_t b = (work + t - 1) / t;
    return (unsigned)(b < cap ? b : cap);
  };

  // zero accumulators: deg + agg1 contiguous, then agg2
  zero_f32<<<blocks((size_t)N * 65, T, 1 << 16), T, 0, stream>>>(deg, (size_t)N * 65);
  zero_f32<<<blocks((size_t)N * 128, T, 1 << 16), T, 0, stream>>>(agg2, (size_t)N * 128);

  // degree, then invert
  degree_kernel<<<blocks(E, T, 1 << 16), T, 0, stream>>>(dstI, deg, E);
  invdeg_kernel<<<(N + T - 1) / T, T, 0, stream>>>(deg, N);

  // ---- layer 1: aggregate x (F=64), GEMM -> h_raw [N,128], BN+ReLU
  {
    size_t total = (size_t)E << 6;
    scatter_feats<<<blocks(total, T, 1 << 17), T, 0, stream>>>(x, srcI, dstI, agg1, total, 6);

    int tiles = ((N + 15) / 16) * (128 / 16);
    sage_gemm_wmma<<<(tiles + 3) / 4, 128, 0, stream>>>(agg1, deg, x, Wl1, Wr1, bl1,
                                                        h, N, 64, 128);

    zero_f32<<<1, 256, 0, stream>>>(stat, 256);
    col_stats<<<512, 128, 0, stream>>>(h, stat, stat + 128, N, 128);
    bn_relu<<<blocks((size_t)N * 128, T, 1 << 16), T, 0, stream>>>(h, h, stat, stat + 128,
                                                                   g1, b1, N, 128);
  }

  // ---- layer 2: aggregate h (F=128), GEMM -> x0raw [N,64], BN+ReLU -> d_out
  {
    size_t total = (size_t)E << 7;
    scatter_feats<<<blocks(total, T, 1 << 17), T, 0, stream>>>(h, srcI, dstI, agg2, total, 7);

    int tiles = ((N + 15) / 16) * (64 / 16);
    sage_gemm_wmma<<<(tiles + 3) / 4, 128, 0, stream>>>(agg2, deg, h, Wl2, Wr2, bl2,
                                                        x0raw, N, 128, 64);

    zero_f32<<<1, 256, 0, stream>>>(stat, 256);
    col_stats<<<512, 64, 0, stream>>>(x0raw, stat, stat + 64, N, 64);
    bn_relu<<<blocks((size_t)N * 64, T, 1 << 16), T, 0, stream>>>(x0raw, out_x0,
                                                                  stat, stat + 64,
                                                                  g2, b2, N, 64);
  }

  // ---- classifier head + softmax -> x1
  fc_softmax<<<(N + T - 1) / T, T, 0, stream>>>(out_x0, Wfc, bfc, out_x1, N);
}